// Interaction_10342281249359
// MI455X (gfx1250) — compile-verified
//
#include <hip/hip_runtime.h>

typedef __attribute__((ext_vector_type(2))) float v2f;
typedef __attribute__((ext_vector_type(8))) float v8f;
typedef int v4i __attribute__((__vector_size__(16)));   // matches builtin's param type

#define NF        26          // sparse features
#define NROWS     27          // 26 sparse + 1 dense
#define DIM       128         // feature dim (K)
#define NPAIR     351         // 27*26/2 upper-triangle entries
#define LSTRIDE   132         // LDS row stride in words (128 + 4 pad, mult of 4)
#define LDS_ROWS  28          // 27 data rows + 1 zero row
#define WAVES_PB  4           // waves (= batches) per block

// CDNA5 async global->LDS copy path (ASYNCcnt), if this toolchain declares it.
#if defined(__has_builtin)
#if __has_builtin(__builtin_amdgcn_global_load_async_to_lds_b128) && \
    __has_builtin(__builtin_amdgcn_s_wait_asynccnt)
#define USE_ASYNC_LDS 1
#endif
#endif

// flat index of (i,j), i<j, in np.triu_indices(27, k=1) row-major order
__device__ __forceinline__ int triu_off(int i, int j) {
    return NF * i - (i * (i - 1)) / 2 + (j - i - 1);
}

__global__ __launch_bounds__(WAVES_PB * 32)
void interact_selfmm_kernel(const float* __restrict__ sparse_x,
                            const float* __restrict__ dense_x,
                            float* __restrict__ out) {
    __shared__ float smem[WAVES_PB * LDS_ROWS * LSTRIDE];

    const int lane   = threadIdx.x & 31;
    const int waveId = threadIdx.x >> 5;
    const int b      = blockIdx.x * WAVES_PB + waveId;

    float* base = smem + waveId * (LDS_ROWS * LSTRIDE);

    // ---- Stage X = [27 x 128] f32 into LDS ----
    // iteration t covers exactly row t: 128 floats = 32 lanes * float4
    #pragma unroll
    for (int t = 0; t < NROWS; ++t) {
        const float* src = (t < NF)
            ? (sparse_x + ((size_t)b * NF + t) * DIM + lane * 4)
            : (dense_x + (size_t)b * DIM + lane * 4);
        float* dst = base + t * LSTRIDE + lane * 4;
#ifdef USE_ASYNC_LDS
        __builtin_amdgcn_global_load_async_to_lds_b128(
            (v4i*)const_cast<float*>(src), (v4i*)dst, 0, 0);
#else
        float4 v = *(const float4*)src;
        *(float4*)dst = v;
#endif
    }
    // zero row 27 (padded rows 27..31 of the 32-row tile read from here)
    for (int w = lane; w < LSTRIDE; w += 32) {
        base[NROWS * LSTRIDE + w] = 0.0f;
    }
#ifdef USE_ASYNC_LDS
    __builtin_amdgcn_s_wait_asynccnt(0);
#endif

    // ---- Fragment base pointers ----
    // A-operand f32 16x4: lane l, VGPR v holds A[m = l&15, k = (l>>4)*2 + v]
    // B-operand (4x16) for B = X^T has the identical per-lane layout, so one
    // b64 LDS load per row-block serves both A and B operands.
    const int m  = lane & 15;
    const int kk = (lane >> 4) * 2;
    int r1 = 16 + m;
    r1 = (r1 > NROWS) ? NROWS : r1;   // rows 28..31 -> zero row 27
    const float* A0p = base + m * LSTRIDE + kk;
    const float* A1p = base + r1 * LSTRIDE + kk;

    v8f c00 = {}; // rows 0-15  x cols 0-15
    v8f c01 = {}; // rows 0-15  x cols 16-31
    v8f c11 = {}; // rows 16-31 x cols 16-31

    #pragma unroll
    for (int k0 = 0; k0 < DIM; k0 += 4) {
        v2f a0 = *(const v2f*)(A0p + k0);
        v2f a1 = *(const v2f*)(A1p + k0);
        c00 = __builtin_amdgcn_wmma_f32_16x16x4_f32(false, a0, false, a0,
                                                    (short)0, c00, false, false);
        c01 = __builtin_amdgcn_wmma_f32_16x16x4_f32(false, a0, false, a1,
                                                    (short)0, c01, false, false);
        c11 = __builtin_amdgcn_wmma_f32_16x16x4_f32(false, a1, false, a1,
                                                    (short)0, c11, false, false);
    }

    // ---- Scatter strict upper triangle ----
    // C/D layout: VGPR r, lane l -> element (M = r + 8*(l>>4), N = l&15)
    float* outb = out + (size_t)b * NPAIR;
    const int hi = lane >> 4;
    const int lo = lane & 15;
    #pragma unroll
    for (int r = 0; r < 8; ++r) {
        const int i = r + 8 * hi;
        // tile (0,0): j = lo
        if (lo > i) outb[triu_off(i, lo)] = c00[r];
        // tile (0,1): j = 16 + lo (always > i); need j <= 26
        if (16 + lo <= NF) outb[triu_off(i, 16 + lo)] = c01[r];
        // tile (1,1): i2 = 16 + i, j2 = 16 + lo
        const int i2 = 16 + i;
        const int j2 = 16 + lo;
        if (i2 < j2 && j2 <= NF) outb[triu_off(i2, j2)] = c11[r];
    }
}

extern "C" void kernel_launch(void* const* d_in, const int* in_sizes, int n_in,
                              void* d_out, int out_size, void* d_ws, size_t ws_size,
                              hipStream_t stream) {
    const float* sparse_x = (const float*)d_in[0]; // [16384, 26, 128] f32
    const float* dense_x  = (const float*)d_in[1]; // [16384, 128] f32
    float* out            = (float*)d_out;         // [16384, 351] f32

    const int B = 16384;
    dim3 block(WAVES_PB * 32);
    dim3 grid(B / WAVES_PB);
    interact_selfmm_kernel<<<grid, block, 0, stream>>>(sparse_x, dense_x, out);
}